// ShiftNMF_32126355374734
// MI455X (gfx1250) — compile-verified
//
#include <hip/hip_runtime.h>
#include <hip/hip_bf16.h>
#include <math.h>

// Problem constants from the reference
#define N_    2048
#define RANK_ 16
#define M_    4096
#define TM    8          // m-values per thread in the modulation kernel

typedef __attribute__((ext_vector_type(2))) float v2f;
typedef __attribute__((ext_vector_type(8))) float v8f;

__device__ __forceinline__ float softplus_f(float x) {
  // log(1+e^x); for large x it is x to fp32 precision
  return (x > 20.0f) ? x : log1pf(__expf(x));
}

// ---------------------------------------------------------------------------
// Kernel 1: softplus(H) -> sH [16 x 4096], softplus(W) -> sW [2048 x 16]
// ---------------------------------------------------------------------------
__global__ void prep_softplus(const float* __restrict__ H,
                              const float* __restrict__ W,
                              float* __restrict__ sH,
                              float* __restrict__ sW) {
  int i = blockIdx.x * blockDim.x + threadIdx.x;
  if (i < RANK_ * M_) sH[i] = softplus_f(H[i]);
  if (i < N_ * RANK_) sW[i] = softplus_f(W[i]);
}

// ---------------------------------------------------------------------------
// Kernel 2: Hft[d,k] = sum_m sH[d,m] * exp(-2*pi*i*m*k/M)   (full DFT)
// Cast as two f32 GEMMs (real & imag) with V_WMMA_F32_16X16X4_F32:
//   D(16d x 16k) += A(16d x 4m) * B(4m x 16k),  K-loop over m in steps of 4.
// A layout (32-bit 16x4): lane L holds rows d=L&15, K slots {kA0, kA0+1},
// kA0 = (L>>4)*2 across its 2 VGPRs. B (4x16) mirrors the same K split, so
// each lane's B rows use the SAME m values as its A slots -> one float2 load
// of sH plus one sincos per B element feeds both matrices.
// One wave per 16-frequency tile; 256 tiles total.
// ---------------------------------------------------------------------------
__global__ void dft_wmma(const float* __restrict__ sH,
                         float2* __restrict__ Hft) {
  const int lane = threadIdx.x & 31;
  const int wave = threadIdx.x >> 5;
  const int kt   = blockIdx.x * (blockDim.x >> 5) + wave;   // 0..255
  const int k0   = kt * 16;

  const int dRow = lane & 15;          // A: matrix row (rank d)
  const int kA0  = (lane >> 4) * 2;    // A/B: K slots {kA0, kA0+1}
  const int kf   = k0 + (lane & 15);   // B: this lane's frequency column

  v8f accR = {0.f,0.f,0.f,0.f,0.f,0.f,0.f,0.f};
  v8f accI = {0.f,0.f,0.f,0.f,0.f,0.f,0.f,0.f};

  const float twoPiOverM = 6.28318530717958647692f / (float)M_;
  const float* aRow = sH + dRow * M_;

  for (int m4 = 0; m4 < M_; m4 += 4) {
    // A fragment: sH[dRow][m4+kA0], sH[dRow][m4+kA0+1]
    v2f a = *(const v2f*)(aRow + m4 + kA0);

    // B fragments: cos / -sin twiddles for the same two m values
    v2f bR, bI;
#pragma unroll
    for (int v = 0; v < 2; ++v) {
      int m  = m4 + kA0 + v;
      int mk = (m * kf) & (M_ - 1);                 // periodic reduction
      float ang = -twoPiOverM * (float)mk;          // -2*pi*(m*k mod M)/M
      float s, c;
      __sincosf(ang, &s, &c);                       // c = cos, s = -sin(2pi mk/M)
      bR[v] = c;
      bI[v] = s;
    }

    accR = __builtin_amdgcn_wmma_f32_16x16x4_f32(
        false, a, false, bR, (short)0, accR, false, false);
    accI = __builtin_amdgcn_wmma_f32_16x16x4_f32(
        false, a, false, bI, (short)0, accI, false, false);
  }

  // D layout (16x16 f32): VGPR r -> row M = r + (lane>=16 ? 8 : 0), N = lane&15
#pragma unroll
  for (int r = 0; r < 8; ++r) {
    int d = r + (lane >> 4) * 8;
    int k = k0 + (lane & 15);
    Hft[d * M_ + k] = make_float2(accR[r], accI[r]);
  }
}

// ---------------------------------------------------------------------------
// Kernel 3: V[n,m] = sum_d sW[n,d] * exp(-2*pi*i*tau[n,d]*m/M) * Hft[d,m]
// Each thread owns TM consecutive m for one n. Per (n,d): 2 sincos (start
// angle + unit rotation), then TM FMA-only complex rotations. Block shares
// one n -> sW/tau loads are wave-uniform (scalar loads).
// ---------------------------------------------------------------------------
__global__ void modulate(const float* __restrict__ sW,
                         const float* __restrict__ tau,
                         const float2* __restrict__ Hft,
                         float2* __restrict__ out) {
  const int n  = blockIdx.y;
  const int m0 = (blockIdx.x * blockDim.x + threadIdx.x) * TM;

  float accR[TM], accI[TM];
#pragma unroll
  for (int j = 0; j < TM; ++j) { accR[j] = 0.f; accI[j] = 0.f; }

  const float negTwoPiOverM = -6.28318530717958647692f / (float)M_;
  const float* swRow  = sW  + n * RANK_;
  const float* tauRow = tau + n * RANK_;

  for (int d = 0; d < RANK_; ++d) {
    float sw = swRow[d];              // uniform across block -> s_load
    float t  = tauRow[d];
    float th0 = negTwoPiOverM * t * (float)m0;   // theta(m0)
    float dth = negTwoPiOverM * t;               // per-step increment
    float s, c, sd, cd;
    __sincosf(th0, &s, &c);
    __sincosf(dth, &sd, &cd);

    const float2* hp = Hft + d * M_ + m0;
#pragma unroll
    for (int j = 0; j < TM; ++j) {
      float2 h = hp[j];
      // sw * (c + i s) * (h.x + i h.y)
      accR[j] = fmaf(sw, fmaf(c, h.x, -s * h.y), accR[j]);
      accI[j] = fmaf(sw, fmaf(c, h.y,  s * h.x), accI[j]);
      // rotate (c,s) by (cd,sd)
      float cn = fmaf(c, cd, -s * sd);
      s        = fmaf(c, sd,  s * cd);
      c        = cn;
    }
  }

  float2* op = out + n * M_ + m0;
#pragma unroll
  for (int j = 0; j < TM; ++j) op[j] = make_float2(accR[j], accI[j]);
}

// ---------------------------------------------------------------------------
extern "C" void kernel_launch(void* const* d_in, const int* in_sizes, int n_in,
                              void* d_out, int out_size, void* d_ws, size_t ws_size,
                              hipStream_t stream) {
  const float* W   = (const float*)d_in[0];   // [2048, 16]
  const float* H   = (const float*)d_in[1];   // [16, 4096]
  const float* tau = (const float*)d_in[2];   // [2048, 16]

  // Workspace layout (all fp32-aligned; < 1 MB total)
  float*  sH  = (float*)d_ws;                               // 16*4096 floats
  float*  sW  = sH + RANK_ * M_;                            // 2048*16 floats
  float2* Hft = (float2*)(sW + N_ * RANK_);                 // 16*4096 float2

  float2* out = (float2*)d_out;   // complex64 interleaved [2048, 4096]

  // 1) softplus prepass (covers both H (64K) and W (32K) elements)
  prep_softplus<<<(RANK_ * M_ + 255) / 256, 256, 0, stream>>>(H, W, sH, sW);

  // 2) DFT of sH via f32 WMMA: 256 k-tiles, 1 wave each, 4 waves/block
  dft_wmma<<<64, 128, 0, stream>>>(sH, Hft);

  // 3) Modulate + rank contraction: one n per blockIdx.y,
  //    M/(TM*256) = 2 blocks along x
  dim3 grid(M_ / (TM * 256), N_, 1);
  modulate<<<grid, 256, 0, stream>>>(sW, tau, Hft, out);
}